// STGCNModel_73924977099265
// MI455X (gfx1250) — compile-verified
//
#include <hip/hip_runtime.h>
#include <hip/hip_bf16.h>

// STGCN: N=50000 nodes, C_IN=5, HID=64, KT=3, E=800000 edges, OUT=1.
// Dominant cost: 2 edge-scatter passes over 192-float node features
// (L2-resident on MI455X's 192MB L2). Dense layer-2 GEMMs run on the
// matrix pipe via V_WMMA_F32_16X16X4_F32 (exact fp32 WMMA).

#define HID 64
#define KT 3
#define CIN 5
#define FEAT (HID * KT)   // 192

typedef __attribute__((ext_vector_type(2))) float v2f;
typedef __attribute__((ext_vector_type(8))) float v8f;

static __device__ __forceinline__ v8f wmma_f32_16x16x4(v2f a, v2f b, v8f c) {
    // 8-arg pattern: (neg_a, A, neg_b, B, c_mod, C, reuse_a, reuse_b)
    return __builtin_amdgcn_wmma_f32_16x16x4_f32(false, a, false, b, (short)0, c,
                                                 false, false);
}

// ---------------- degree / dinv ----------------
__global__ void deg_init(float* deg, int N) {
    int i = blockIdx.x * blockDim.x + threadIdx.x;
    if (i < N) deg[i] = 1.0f;   // self-loop contributes 1
}

__global__ void deg_acc(const int* __restrict__ dst, float* deg, int E) {
    int e = blockIdx.x * blockDim.x + threadIdx.x;
    if (e < E) atomicAdd(&deg[dst[e]], 1.0f);
}

__global__ void deg_to_dinv(float* deg, int N) {
    int i = blockIdx.x * blockDim.x + threadIdx.x;
    if (i < N) deg[i] = rsqrtf(fmaxf(deg[i], 1.0f));
}

// ---------------- layer 1 dense (K=5 too small for WMMA) ----------------
// hw1[n,o,t] = sum_c x[n,c,t] * W1[c,o]
// acc1[n,o,t] = dinv[n]^2 * hw1[n,o,t] + tp[n,o] + b1[o] + bt1[o]  (self-loop folded)
__global__ __launch_bounds__(256) void layer1_dense(
    const float* __restrict__ x, const float* __restrict__ W1,
    const float* __restrict__ b1, const float* __restrict__ Wt1,
    const float* __restrict__ bt1, const float* __restrict__ dinv,
    float* __restrict__ hw1, float* __restrict__ acc1, int N)
{
    int gid = blockIdx.x * blockDim.x + threadIdx.x;
    if (gid >= N * HID) return;
    int n = gid >> 6;
    int o = gid & 63;
    const float* xr = x + (size_t)n * (CIN * KT);
    float hw[KT] = {0.f, 0.f, 0.f};
    float tp = 0.f;
#pragma unroll
    for (int c = 0; c < CIN; ++c) {
        float w = W1[c * HID + o];
#pragma unroll
        for (int t = 0; t < KT; ++t) {
            float xv = xr[c * KT + t];
            hw[t] += xv * w;
            tp += xv * Wt1[o * (CIN * KT) + c * KT + t];
        }
    }
    float di = dinv[n];
    float selfn = di * di;
    float base = tp + b1[o] + bt1[o];
    size_t idx = (size_t)n * FEAT + o * KT;
#pragma unroll
    for (int t = 0; t < KT; ++t) {
        hw1[idx + t]  = hw[t];
        acc1[idx + t] = selfn * hw[t] + base;
    }
}

// ---------------- edge scatter (both layers) ----------------
// acc[dst] += dinv[src]*dinv[dst] * feat[src]; 48 threads/edge, float4 gather,
// f32 atomic adds land in L2 (38.4MB accumulator is L2-resident).
__global__ __launch_bounds__(256) void scatter_edges(
    const int* __restrict__ src, const int* __restrict__ dst,
    const float* __restrict__ dinv, const float* __restrict__ feat,
    float* acc, int E)
{
    int gid = blockIdx.x * blockDim.x + threadIdx.x;
    int e = gid / 48;
    int j = gid % 48;
    if (e >= E) return;
    int s = src[e];
    int d = dst[e];
    float nrm = dinv[s] * dinv[d];
    const float4 v = ((const float4*)(feat + (size_t)s * FEAT))[j];
    float* ap = acc + (size_t)d * FEAT + j * 4;
    atomicAdd(ap + 0, nrm * v.x);
    atomicAdd(ap + 1, nrm * v.y);
    atomicAdd(ap + 2, nrm * v.z);
    atomicAdd(ap + 3, nrm * v.w);
}

// ---------------- layer 2 dense: fp32 WMMA ----------------
// Input: pre-relu layer-1 accumulator hin [N,192] (relu fused into A loads).
// Per wave: 16-node tile x 16-output-channel block.
//   hw2[n,o,t] = sum_c relu(h[n,c,t]) * W2[c,o]          (3 accs, K=64)
//   tp2[n,o]   = sum_k relu(h[n,k])   * Wt2[o,k]         (1 acc,  K=192)
// A 16x4 layout: lane m=lane&15 holds K = (lane>>4)*2 + {0,1}
// B 4x16 layout: lane col n=lane&15 holds K rows (lane>>4)*2 + {0,1}
// D 16x16: VGPR j -> row j + (lane>=16 ? 8 : 0), col = lane&15
__global__ __launch_bounds__(128) void layer2_dense_wmma(
    const float* __restrict__ hin, const float* __restrict__ W2,
    const float* __restrict__ Wt2, const float* __restrict__ b2,
    const float* __restrict__ bt2,
    float* __restrict__ hw_out, float* __restrict__ tpb_out, int N)
{
    const int wave = threadIdx.x >> 5;          // o-block 0..3
    const int lane = threadIdx.x & 31;
    const int n0   = blockIdx.x * 16;           // node tile base
    const int m    = lane & 15;
    const int kh   = lane >> 4;                 // K half-select
    int node = n0 + m;
    if (node >= N) node = N - 1;                // tail clamp (loads only)
    const int o = wave * 16 + m;                // B column / store column
    const float* hrow = hin + (size_t)node * FEAT;

    v8f acc_t0 = {}; v8f acc_t1 = {}; v8f acc_t2 = {}; v8f acc_tp = {};

    // hw2: K over channels, step 4
    for (int cc = 0; cc < HID; cc += 4) {
        const int c0 = cc + kh * 2;
        v2f b;
        b.x = W2[(c0 + 0) * HID + o];
        b.y = W2[(c0 + 1) * HID + o];
        v2f a;
        a.x = fmaxf(hrow[(c0 + 0) * KT + 0], 0.f);
        a.y = fmaxf(hrow[(c0 + 1) * KT + 0], 0.f);
        acc_t0 = wmma_f32_16x16x4(a, b, acc_t0);
        a.x = fmaxf(hrow[(c0 + 0) * KT + 1], 0.f);
        a.y = fmaxf(hrow[(c0 + 1) * KT + 1], 0.f);
        acc_t1 = wmma_f32_16x16x4(a, b, acc_t1);
        a.x = fmaxf(hrow[(c0 + 0) * KT + 2], 0.f);
        a.y = fmaxf(hrow[(c0 + 1) * KT + 2], 0.f);
        acc_t2 = wmma_f32_16x16x4(a, b, acc_t2);
    }

    // tp2: K over flattened (c,t) = 192, step 4
    for (int k = 0; k < FEAT; k += 4) {
        const int k0 = k + kh * 2;
        v2f a;
        a.x = fmaxf(hrow[k0 + 0], 0.f);
        a.y = fmaxf(hrow[k0 + 1], 0.f);
        v2f b;
        b.x = Wt2[o * FEAT + k0 + 0];
        b.y = Wt2[o * FEAT + k0 + 1];
        acc_tp = wmma_f32_16x16x4(a, b, acc_tp);
    }

    const float bias = b2[o] + bt2[o];
#pragma unroll
    for (int j = 0; j < 8; ++j) {
        int nd = n0 + j + kh * 8;
        if (nd >= N) continue;
        size_t base = (size_t)nd * FEAT + o * KT;
        hw_out[base + 0] = acc_t0[j];
        hw_out[base + 1] = acc_t1[j];
        hw_out[base + 2] = acc_t2[j];
        tpb_out[(size_t)nd * HID + o] = acc_tp[j] + bias;
    }
}

// ---------------- layer 2 accumulator init ----------------
__global__ __launch_bounds__(256) void init_acc2(
    const float* __restrict__ hw2, const float* __restrict__ tpb2,
    const float* __restrict__ dinv, float* __restrict__ acc2, int N)
{
    int gid = blockIdx.x * blockDim.x + threadIdx.x;
    if (gid >= N * HID) return;
    int n = gid >> 6;
    int o = gid & 63;
    float di = dinv[n];
    float selfn = di * di;
    float tb = tpb2[gid];
    size_t idx = (size_t)n * FEAT + o * KT;
#pragma unroll
    for (int t = 0; t < KT; ++t)
        acc2[idx + t] = selfn * hw2[idx + t] + tb;
}

// ---------------- final FC (relu fused) ----------------
__global__ __launch_bounds__(256) void final_fc(
    const float* __restrict__ acc2, const float* __restrict__ Wf,
    const float* __restrict__ bf, float* __restrict__ out, int N)
{
    int gid = blockIdx.x * blockDim.x + threadIdx.x;
    if (gid >= N * KT) return;
    int n = gid / KT;
    int t = gid % KT;
    const float* h = acc2 + (size_t)n * FEAT + t;
    float s = 0.f;
#pragma unroll
    for (int o = 0; o < HID; ++o)
        s += fmaxf(h[o * KT], 0.f) * Wf[o];
    out[gid] = s + bf[0];
}

extern "C" void kernel_launch(void* const* d_in, const int* in_sizes, int n_in,
                              void* d_out, int out_size, void* d_ws, size_t ws_size,
                              hipStream_t stream)
{
    const float* x   = (const float*)d_in[0];
    const int*  eidx = (const int*) d_in[1];
    const float* W1  = (const float*)d_in[2];
    const float* b1  = (const float*)d_in[3];
    const float* Wt1 = (const float*)d_in[4];
    const float* bt1 = (const float*)d_in[5];
    const float* W2  = (const float*)d_in[6];
    const float* b2  = (const float*)d_in[7];
    const float* Wt2 = (const float*)d_in[8];
    const float* bt2 = (const float*)d_in[9];
    const float* Wf  = (const float*)d_in[10];
    const float* bf  = (const float*)d_in[11];
    float* out = (float*)d_out;

    const int N = in_sizes[0] / (CIN * KT);
    const int E = in_sizes[1] / 2;
    const int* srcI = eidx;
    const int* dstI = eidx + E;

    // Workspace layout (f32): deg/dinv[N] | bufA[N*192] | bufB[N*64] | bufC[N*192]
    float* deg  = (float*)d_ws;
    float* bufA = deg  + N;                    // hw features (both layers)
    float* bufB = bufA + (size_t)N * FEAT;     // tp + bias (layer 2)
    float* bufC = bufB + (size_t)N * HID;      // scatter accumulator (both layers)

    const int TB = 256;
    // degree -> dinv (in place)
    deg_init   <<<(N + TB - 1) / TB, TB, 0, stream>>>(deg, N);
    deg_acc    <<<(E + TB - 1) / TB, TB, 0, stream>>>(dstI, deg, E);
    deg_to_dinv<<<(N + TB - 1) / TB, TB, 0, stream>>>(deg, N);

    // layer 1: dense + folded self-loop/bias init, then edge scatter
    layer1_dense<<<(N * HID + TB - 1) / TB, TB, 0, stream>>>(
        x, W1, b1, Wt1, bt1, deg, bufA, bufC, N);
    {
        long long tot = (long long)E * 48;
        scatter_edges<<<(int)((tot + TB - 1) / TB), TB, 0, stream>>>(
            srcI, dstI, deg, bufA, bufC, E);
    }

    // layer 2: WMMA dense (relu fused on input), init, scatter
    layer2_dense_wmma<<<(N + 15) / 16, 128, 0, stream>>>(
        bufC, W2, Wt2, b2, bt2, bufA, bufB, N);
    init_acc2<<<(N * HID + TB - 1) / TB, TB, 0, stream>>>(bufA, bufB, deg, bufC, N);
    {
        long long tot = (long long)E * 48;
        scatter_edges<<<(int)((tot + TB - 1) / TB), TB, 0, stream>>>(
            srcI, dstI, deg, bufA, bufC, E);
    }

    // final FC with fused relu
    final_fc<<<(N * KT + TB - 1) / TB, TB, 0, stream>>>(bufC, Wf, bf, out, N);
}